// ExplicitGCN_5557687681111
// MI455X (gfx1250) — compile-verified
//
#include <hip/hip_runtime.h>
#include <hip/hip_bf16.h>

// Problem constants (match reference).
#define V_N     10000
#define E_N     60000
#define B_N     8
#define LATENT  512
#define HIDDEN  256
#define NLAYERS 5
#define NNODES  (B_N * V_N)          // 80000 rows, divisible by 64

typedef float v2f __attribute__((ext_vector_type(2)));
typedef float v8f __attribute__((ext_vector_type(8)));

#if defined(__gfx1250__) && __has_builtin(__builtin_amdgcn_wmma_f32_16x16x4_f32)
#define HAVE_WMMA_F32 1
#endif

// ---------------- GCN normalization precompute ----------------

__global__ void k_deg_init(float* deg) {
    int v = blockIdx.x * blockDim.x + threadIdx.x;
    if (v < V_N) deg[v] = 1.0f;      // self-loop
}

__global__ void k_deg_acc(float* deg, const int* __restrict__ dst) {
    int e = blockIdx.x * blockDim.x + threadIdx.x;
    if (e < E_N) atomicAdd(&deg[dst[e]], 1.0f);
}

__global__ void k_dinv(float* deg_to_dinv, float* selfnorm) {
    int v = blockIdx.x * blockDim.x + threadIdx.x;
    if (v < V_N) {
        float di = rsqrtf(deg_to_dinv[v]);   // deg >= 1 due to self-loop
        deg_to_dinv[v] = di;
        selfnorm[v] = di * di;
    }
}

__global__ void k_norm(const float* __restrict__ dinv, const int* __restrict__ src,
                       const int* __restrict__ dst, float* __restrict__ norm) {
    int e = blockIdx.x * blockDim.x + threadIdx.x;
    if (e < E_N) norm[e] = dinv[src[e]] * dinv[dst[e]];
}

// ---------------- Input layer (factored through the concat) ----------------

// xyzW[v,h] = input_b[h] + sum_{i<3} xyz[v,i] * W[h,i]
__global__ void k_xyzW(const float* __restrict__ xyz, const float* __restrict__ iW,
                       const float* __restrict__ ib, float* __restrict__ xyzW) {
    int i = blockIdx.x * blockDim.x + threadIdx.x;   // V_N * HIDDEN
    if (i >= V_N * HIDDEN) return;
    int v = i / HIDDEN, hh = i % HIDDEN;
    const float* wp = iW + (size_t)hh * (LATENT + 3);
    xyzW[i] = ib[hh] + xyz[3 * v] * wp[0] + xyz[3 * v + 1] * wp[1] + xyz[3 * v + 2] * wp[2];
}

// latW[b,h] = sum_{j<512} latent[b,j] * W[h, 3+j]
__global__ void k_latW(const float* __restrict__ latent, const float* __restrict__ iW,
                       float* __restrict__ latW) {
    int i = blockIdx.x * blockDim.x + threadIdx.x;   // B_N * HIDDEN
    if (i >= B_N * HIDDEN) return;
    int b = i / HIDDEN, hh = i % HIDDEN;
    const float* lp = latent + (size_t)b * LATENT;
    const float* wp = iW + (size_t)hh * (LATENT + 3) + 3;
    float s = 0.0f;
    for (int j = 0; j < LATENT; ++j) s += lp[j] * wp[j];
    latW[i] = s;
}

// x0[b,v,h] = relu(xyzW[v,h] + latW[b,h])
__global__ void k_x0(const float* __restrict__ xyzW, const float* __restrict__ latW,
                     float* __restrict__ x) {
    size_t i = (size_t)blockIdx.x * blockDim.x + threadIdx.x;  // NNODES*HIDDEN
    int hh = (int)(i & (HIDDEN - 1));
    size_t node = i >> 8;
    int v = (int)(node % V_N);
    int b = (int)(node / V_N);
    float s = xyzW[(size_t)v * HIDDEN + hh] + latW[b * HIDDEN + hh];
    x[i] = s > 0.0f ? s : 0.0f;
}

// ---------------- fp32 WMMA GEMM: h[node, o] = sum_k x[node, k] * W[o, k] ----------------
// Register-blocked: each wave owns a 32(M) x 64(N) tile = 2x4 grid of 16x16 WMMA tiles.
// Per K-step (K+=4): 2 A-frag + 4 B-frag loads feed 8 V_WMMA_F32_16X16X4_F32 ops.
// Block = 8 waves -> 64M x 256N (full N); grid (NNODES/64). No divergence (EXEC all ones).

#define GEMM_MT 2
#define GEMM_NT 4

__global__ void __launch_bounds__(256)
k_gemm_wmma(const float* __restrict__ x, const float* __restrict__ W,
            float* __restrict__ h) {
    const int lane  = threadIdx.x & 31;
    const int wv    = threadIdx.x >> 5;                 // 0..7
    const int half  = lane >> 4;                        // 0 -> K{0,1}, 1 -> K{2,3}
    const int l16   = lane & 15;
    const int mbase = blockIdx.x * 64 + (wv >> 2) * 32; // 2 M-groups of 32 rows
    const int nbase = (wv & 3) * 64;                    // 4 N-groups of 64 cols

    const float* arow = x + (size_t)(mbase + l16) * HIDDEN + half * 2;  // A(m, k..k+1)
    const float* bcol = W + (size_t)(nbase + l16) * HIDDEN + half * 2;  // B(k..k+1, n)=W[n][k]

    v8f acc[GEMM_MT][GEMM_NT] = {};

#if HAVE_WMMA_F32
#pragma unroll 4
    for (int k = 0; k < HIDDEN; k += 4) {
        v2f a[GEMM_MT], b[GEMM_NT];
#pragma unroll
        for (int mt = 0; mt < GEMM_MT; ++mt)
            a[mt] = *(const v2f*)(arow + (size_t)mt * 16 * HIDDEN + k);
#pragma unroll
        for (int nt = 0; nt < GEMM_NT; ++nt)
            b[nt] = *(const v2f*)(bcol + (size_t)nt * 16 * HIDDEN + k);
#pragma unroll
        for (int mt = 0; mt < GEMM_MT; ++mt)
#pragma unroll
            for (int nt = 0; nt < GEMM_NT; ++nt)
                acc[mt][nt] = __builtin_amdgcn_wmma_f32_16x16x4_f32(
                    /*neg_a=*/false, a[mt], /*neg_b=*/false, b[nt],
                    /*c_mod=*/(short)0, acc[mt][nt],
                    /*reuse_a=*/false, /*reuse_b=*/false);
    }
#else
    // Scalar fallback with identical C-layout: acc[mt][nt][r] = D(m = mt*16 + 8*half + r,
    //                                                            n = nt*16 + l16)
#pragma unroll
    for (int mt = 0; mt < GEMM_MT; ++mt)
#pragma unroll
        for (int r = 0; r < 8; ++r) {
            const float* xr = x + (size_t)(mbase + mt * 16 + half * 8 + r) * HIDDEN;
#pragma unroll
            for (int nt = 0; nt < GEMM_NT; ++nt) {
                const float* wr = W + (size_t)(nbase + nt * 16 + l16) * HIDDEN;
                float s = 0.0f;
                for (int k = 0; k < HIDDEN; ++k) s += xr[k] * wr[k];
                acc[mt][nt][r] = s;
            }
        }
#endif

    // C/D layout: VGPR r -> (M = mt*16 + 8*half + r, N = nt*16 + l16)
    float* outp = h + (size_t)(mbase + half * 8) * HIDDEN + (nbase + l16);
#pragma unroll
    for (int mt = 0; mt < GEMM_MT; ++mt)
#pragma unroll
        for (int nt = 0; nt < GEMM_NT; ++nt)
#pragma unroll
            for (int r = 0; r < 8; ++r)
                outp[((size_t)mt * 16 + r) * HIDDEN + nt * 16] = acc[mt][nt][r];
}

// ---------------- Message passing ----------------

// agg[b,v,:] = h[b,v,:] * selfnorm[v]   (initializes the accumulator deterministically)
__global__ void k_selfinit(const float* __restrict__ h, float* __restrict__ agg,
                           const float* __restrict__ selfnorm) {
    size_t i = (size_t)blockIdx.x * blockDim.x + threadIdx.x;  // NNODES*HIDDEN
    int v = (int)((i >> 8) % V_N);
    agg[i] = h[i] * selfnorm[v];
}

// agg[b,dst,:] += h[b,src,:] * norm[e]  — one thread per (edge, 4-feature chunk), per batch.
__global__ void k_scatter(const float* __restrict__ h, float* __restrict__ agg,
                          const int* __restrict__ src, const int* __restrict__ dst,
                          const float* __restrict__ norm) {
    int t  = blockIdx.x * blockDim.x + threadIdx.x;  // E_N * 64
    int e  = t >> 6;
    int f4 = (t & 63) << 2;
    int b  = blockIdx.y;
    int s  = src[e], d = dst[e];
    float w = norm[e];
    const float* hp = h + ((size_t)b * V_N + s) * HIDDEN + f4;
    float*       ap = agg + ((size_t)b * V_N + d) * HIDDEN + f4;
    float4 hv = *(const float4*)hp;
    atomicAdd(ap + 0, hv.x * w);
    atomicAdd(ap + 1, hv.y * w);
    atomicAdd(ap + 2, hv.z * w);
    atomicAdd(ap + 3, hv.w * w);
}

// x = relu(agg + bias) in-place
__global__ void k_relu_bias(float* __restrict__ x, const float* __restrict__ bias) {
    size_t i = (size_t)blockIdx.x * blockDim.x + threadIdx.x;  // NNODES*HIDDEN
    float s = x[i] + bias[i & (HIDDEN - 1)];
    x[i] = s > 0.0f ? s : 0.0f;
}

// ---------------- Output projection: disp[node, o] = x[node,:] . out_W[o,:] + out_b[o]
__global__ void k_out(const float* __restrict__ x, const float* __restrict__ oW,
                      const float* __restrict__ ob, float* __restrict__ disp) {
    int idx = blockIdx.x * blockDim.x + threadIdx.x;
    if (idx >= NNODES * 3) return;
    int node = idx / 3, o = idx % 3;
    const float* xp = x + (size_t)node * HIDDEN;
    const float* wp = oW + o * HIDDEN;
    float s = ob[o];
#pragma unroll 4
    for (int k = 0; k < HIDDEN; k += 4) {
        float4 xv = *(const float4*)(xp + k);
        float4 wv = *(const float4*)(wp + k);
        s += xv.x * wv.x + xv.y * wv.y + xv.z * wv.z + xv.w * wv.w;
    }
    disp[idx] = s;
}

// ---------------- Host launch ----------------

extern "C" void kernel_launch(void* const* d_in, const int* in_sizes, int n_in,
                              void* d_out, int out_size, void* d_ws, size_t ws_size,
                              hipStream_t stream) {
    const float* vertex_xyz = (const float*)d_in[0];   // [V,3]
    const float* latent     = (const float*)d_in[1];   // [B,512]
    const int*   edge_index = (const int*)d_in[2];     // [2,E]
    const float* input_W    = (const float*)d_in[3];   // [256,515]
    const float* input_b    = (const float*)d_in[4];   // [256]
    const float* conv_W     = (const float*)d_in[5];   // [5,256,256]
    const float* conv_b     = (const float*)d_in[6];   // [5,256]
    const float* out_W      = (const float*)d_in[7];   // [3,256]
    const float* out_b      = (const float*)d_in[8];   // [3]
    float* disp = (float*)d_out;                       // [B,V,3]

    const int* src = edge_index;
    const int* dst = edge_index + E_N;

    // Workspace carve (floats). Total ~175 MB.
    float* xbuf     = (float*)d_ws;                         // [NNODES, 256] activations / agg
    float* hbuf     = xbuf + (size_t)NNODES * HIDDEN;       // [NNODES, 256] linear output
    float* xyzW     = hbuf + (size_t)NNODES * HIDDEN;       // [V, 256]
    float* latW     = xyzW + (size_t)V_N * HIDDEN;          // [B, 256]
    float* dinv     = latW + (size_t)B_N * HIDDEN;          // [V] (deg, then rsqrt(deg))
    float* selfnorm = dinv + V_N;                           // [V]
    float* norm     = selfnorm + V_N;                       // [E]

    const int T = 256;
    const size_t NH = (size_t)NNODES * HIDDEN;              // 20.48M

    // 1) GCN normalization
    k_deg_init<<<(V_N + T - 1) / T, T, 0, stream>>>(dinv);
    k_deg_acc<<<(E_N + T - 1) / T, T, 0, stream>>>(dinv, dst);
    k_dinv<<<(V_N + T - 1) / T, T, 0, stream>>>(dinv, selfnorm);
    k_norm<<<(E_N + T - 1) / T, T, 0, stream>>>(dinv, src, dst, norm);

    // 2) Input layer (factored)
    k_xyzW<<<(V_N * HIDDEN + T - 1) / T, T, 0, stream>>>(vertex_xyz, input_W, input_b, xyzW);
    k_latW<<<(B_N * HIDDEN + T - 1) / T, T, 0, stream>>>(latent, input_W, latW);
    k_x0<<<(unsigned)(NH / T), T, 0, stream>>>(xyzW, latW, xbuf);

    // 3) GCN layers
    dim3 ggrid(NNODES / 64);                                // 1250 blocks, 8 waves each
    dim3 sgrid((E_N * 64) / T, B_N);                        // (15000, 8)
    for (int l = 0; l < NLAYERS; ++l) {
        k_gemm_wmma<<<ggrid, T, 0, stream>>>(xbuf, conv_W + (size_t)l * HIDDEN * HIDDEN, hbuf);
        k_selfinit<<<(unsigned)(NH / T), T, 0, stream>>>(hbuf, xbuf, selfnorm);
        k_scatter<<<sgrid, T, 0, stream>>>(hbuf, xbuf, src, dst, norm);
        k_relu_bias<<<(unsigned)(NH / T), T, 0, stream>>>(xbuf, conv_b + (size_t)l * HIDDEN);
    }

    // 4) Output projection
    k_out<<<(NNODES * 3 + T - 1) / T, T, 0, stream>>>(xbuf, out_W, out_b, disp);
}